// WESUP_6193342841588
// MI455X (gfx1250) — compile-verified
//
#include <hip/hip_runtime.h>
#include <hip/hip_bf16.h>
#include <math.h>

// ---------------- problem constants (match reference) ----------------
#define SEGS   440
#define SEGPAD 448          // 28 tiles of 16
#define CFEAT  2112
#define HWPIX  (256*256)
#define H1DIM  1024
#define H3DIM  32

typedef float v2f __attribute__((ext_vector_type(2)));
typedef float v8f __attribute__((ext_vector_type(8)));

// ---------------- ws layout (float offsets) ----------------
#define OFF_SUMS   0                         // SEGPAD*CFEAT = 946176 (sums -> feat in place)
#define OFF_COUNTS 946176                    // SEGPAD
#define OFF_HIST   946624                    // SEGPAD*3
#define OFF_FN     947968                    // SEGPAD*CFEAT
#define OFF_H1     1894144                   // SEGPAD*1024
#define OFF_H2     2352896                   // SEGPAD*1024
#define OFF_H3     2811648                   // SEGPAD*32
#define OFF_AFF    2825984                   // SEGPAD*SEGPAD = 200704
#define OFF_LAB    3026688                   // SEGPAD ints

// ---------------- zero init of accumulators ----------------
__global__ void WESUP_zero(float* __restrict__ p, int n) {
    int i = blockIdx.x * blockDim.x + threadIdx.x;
    if (i < n) p[i] = 0.0f;
}

// ---------------- segment sums: LDS-accumulated streaming pass ----------------
// grid = (CFEAT/16) * PCHUNKS blocks, 256 threads.
#define CPB 16
#define PCHUNKS 8
__global__ void WESUP_segsum(const float* __restrict__ fm, const int* __restrict__ sp,
                             float* __restrict__ sums) {
    __shared__ float lsum[SEGS * CPB];          // 28160 B
    for (int i = threadIdx.x; i < SEGS * CPB; i += blockDim.x) lsum[i] = 0.0f;
    __syncthreads();

    const int ncblk  = CFEAT / CPB;             // 132
    const int cblk   = blockIdx.x % ncblk;
    const int pblk   = blockIdx.x / ncblk;
    const int clocal = threadIdx.x >> 4;        // 0..15
    const int poff   = threadIdx.x & 15;
    const int c      = cblk * CPB + clocal;
    const float* fmc = fm + (size_t)c * HWPIX;

    const int chunk = HWPIX / PCHUNKS;          // 8192
    const int p0    = pblk * chunk;
    for (int p = p0 + poff; p < p0 + chunk; p += 16) {
        float v = fmc[p];
        int   s = sp[p];
        atomicAdd(&lsum[s * CPB + clocal], v);  // ds_add_f32
    }
    __syncthreads();
    for (int i = threadIdx.x; i < SEGS * CPB; i += blockDim.x) {
        int s = i / CPB, cl = i % CPB;
        float v = lsum[i];
        if (v != 0.0f)
            atomicAdd(&sums[(size_t)s * CFEAT + cblk * CPB + cl], v);
    }
}

// ---------------- pixel counts + class histogram ----------------
__global__ void WESUP_counthist(const int* __restrict__ sp, const int* __restrict__ y,
                                float* __restrict__ counts, float* __restrict__ hist) {
    int p = blockIdx.x * blockDim.x + threadIdx.x;
    if (p >= HWPIX) return;
    int s = sp[p];
    atomicAdd(&counts[s], 1.0f);
    atomicAdd(&hist[s * 3 + y[p]], 1.0f);
}

// ---------------- feat = sums/max(count,1) (in place) ; fn = feat/||feat|| ----------------
__global__ void WESUP_featnorm(float* __restrict__ feat, const float* __restrict__ counts,
                               float* __restrict__ fn) {
    int s = blockIdx.x;                         // 0..SEGS-1
    float inv = 1.0f / fmaxf(counts[s], 1.0f);
    float* row = feat + (size_t)s * CFEAT;
    float ss = 0.0f;
    for (int c = threadIdx.x; c < CFEAT; c += blockDim.x) {
        float v = row[c] * inv;
        row[c] = v;
        ss += v * v;
    }
    __shared__ float red[256];
    red[threadIdx.x] = ss;
    __syncthreads();
    for (int o = 128; o > 0; o >>= 1) {
        if (threadIdx.x < o) red[threadIdx.x] += red[threadIdx.x + o];
        __syncthreads();
    }
    float rinv = 1.0f / sqrtf(red[0]);          // reference has no guard either
    for (int c = threadIdx.x; c < CFEAT; c += blockDim.x)
        fn[(size_t)s * CFEAT + c] = row[c] * rinv;
}

// ---------------- f32 WMMA GEMM: one wave per 16x16 tile, K step 4 ----------------
// mode 0: C = relu(A@B + bias)     (B row-major KxN, ldb = Nout)
// mode 1: C = exp(-(A@B))          (btrans: B is Nout x K row-major, B^T used)
__global__ void WESUP_gemm_wmma(const float* __restrict__ A, const float* __restrict__ Bm,
                                const float* __restrict__ bias, float* __restrict__ Cm,
                                int Mtiles, int Ntiles, int K, int mode, int btrans) {
    int wid = (int)((blockIdx.x * blockDim.x + threadIdx.x) >> 5);
    if (wid >= Mtiles * Ntiles) return;
    int lane = threadIdx.x & 31;
    int tm = wid / Ntiles, tn = wid % Ntiles;
    int Nout = Ntiles * 16;
    int l15  = lane & 15;
    int kgrp = lane >> 4;                        // 0 or 1

    const float* Arow = A + (size_t)(tm * 16 + l15) * K;
    v8f acc = {};
    if (!btrans) {
        const float* Bcol = Bm + (tn * 16 + l15);
        for (int k = 0; k < K; k += 4) {
            int ka = k + 2 * kgrp;
            v2f a, b;
            a[0] = Arow[ka];
            a[1] = Arow[ka + 1];
            b[0] = Bcol[(size_t)ka * Nout];
            b[1] = Bcol[(size_t)(ka + 1) * Nout];
            acc = __builtin_amdgcn_wmma_f32_16x16x4_f32(false, a, false, b,
                                                        (short)0, acc, false, false);
        }
    } else {
        const float* Brow = Bm + (size_t)(tn * 16 + l15) * K;
        for (int k = 0; k < K; k += 4) {
            int ka = k + 2 * kgrp;
            v2f a, b;
            a[0] = Arow[ka];
            a[1] = Arow[ka + 1];
            b[0] = Brow[ka];
            b[1] = Brow[ka + 1];
            acc = __builtin_amdgcn_wmma_f32_16x16x4_f32(false, a, false, b,
                                                        (short)0, acc, false, false);
        }
    }
    // D layout: VGPR i -> row tm*16 + i (+8 for lanes 16..31), col tn*16 + (lane&15)
    int col  = tn * 16 + l15;
    int row0 = tm * 16 + (kgrp ? 8 : 0);
    float bb = bias ? bias[col] : 0.0f;
#pragma unroll
    for (int i = 0; i < 8; ++i) {
        float v = acc[i];
        v = (mode == 0) ? fmaxf(v + bb, 0.0f) : expf(-v);
        Cm[(size_t)(row0 + i) * Nout + col] = v;
    }
}

// ---------------- classifier head: softmax(h3 @ Wc + bc) ----------------
__global__ void WESUP_classifier(const float* __restrict__ h3, const float* __restrict__ Wc,
                                 const float* __restrict__ bc, float* __restrict__ out) {
    int s = blockIdx.x * blockDim.x + threadIdx.x;
    if (s >= SEGS) return;
    float z0 = bc[0], z1 = bc[1];
    for (int d = 0; d < H3DIM; ++d) {
        float v = h3[(size_t)s * H3DIM + d];
        z0 += v * Wc[d * 2 + 0];
        z1 += v * Wc[d * 2 + 1];
    }
    float mx = fmaxf(z0, z1);
    float e0 = expf(z0 - mx), e1 = expf(z1 - mx);
    float inv = 1.0f / (e0 + e1);
    out[s * 2 + 0] = e0 * inv;
    out[s * 2 + 1] = e1 * inv;
}

// ---------------- label from histogram with tie-break ----------------
__global__ void WESUP_labels(const float* __restrict__ hist, int* __restrict__ lab) {
    int s = blockIdx.x * blockDim.x + threadIdx.x;
    if (s >= SEGS) return;
    float h0 = hist[s * 3 + 0], h1 = hist[s * 3 + 1], h2 = hist[s * 3 + 2];
    int l = 0; float m = h0;                    // argmax, first index wins ties
    if (h1 > m) { m = h1; l = 1; }
    if (h2 > m) { m = h2; l = 2; }
    if (l == 0) l = (h2 > h1) ? 2 : ((h1 > h2) ? 1 : 0);
    lab[s] = l;
}

// ---------------- row max/argmax over masked affinity -> new labels ----------------
__global__ void WESUP_newlab(const float* __restrict__ aff, const int* __restrict__ lab,
                             float* __restrict__ outl) {
    int i = blockIdx.x;                          // 0..SEGS-1
    __shared__ float bv[256];
    __shared__ int   bi[256];
    float best = -INFINITY;
    int   bidx = 0;
    for (int j = threadIdx.x; j < SEGS; j += 256) {
        if (lab[j] != 0) {
            float v = aff[(size_t)i * SEGPAD + j];
            if (v > best) { best = v; bidx = j; }   // ascending j -> first max kept
        }
    }
    bv[threadIdx.x] = best;
    bi[threadIdx.x] = bidx;
    __syncthreads();
    for (int o = 128; o > 0; o >>= 1) {
        if (threadIdx.x < o) {
            float vo = bv[threadIdx.x + o];
            int   io = bi[threadIdx.x + o];
            if (vo > bv[threadIdx.x] ||
                (vo == bv[threadIdx.x] && io < bi[threadIdx.x])) {
                bv[threadIdx.x] = vo;
                bi[threadIdx.x] = io;
            }
        }
        __syncthreads();
    }
    if (threadIdx.x == 0) {
        int li = lab[i];
        int nl = li;
        if (li == 0 && bv[0] >= 0.7f) nl = lab[bi[0]];
        outl[i] = (float)nl;
    }
}

// ---------------- launcher ----------------
extern "C" void kernel_launch(void* const* d_in, const int* in_sizes, int n_in,
                              void* d_out, int out_size, void* d_ws, size_t ws_size,
                              hipStream_t stream) {
    const float* fm = (const float*)d_in[0];
    const int*   sp = (const int*)d_in[1];
    const int*   y  = (const int*)d_in[2];
    const float* W1 = (const float*)d_in[3];
    const float* b1 = (const float*)d_in[4];
    const float* W2 = (const float*)d_in[5];
    const float* b2 = (const float*)d_in[6];
    const float* W3 = (const float*)d_in[7];
    const float* b3 = (const float*)d_in[8];
    const float* Wc = (const float*)d_in[9];
    const float* bc = (const float*)d_in[10];

    float* out = (float*)d_out;
    float* ws  = (float*)d_ws;

    float* sums   = ws + OFF_SUMS;     // -> feat in place
    float* counts = ws + OFF_COUNTS;
    float* hist   = ws + OFF_HIST;
    float* fn     = ws + OFF_FN;
    float* h1     = ws + OFF_H1;
    float* h2     = ws + OFF_H2;
    float* h3     = ws + OFF_H3;
    float* aff    = ws + OFF_AFF;
    int*   lab    = (int*)(ws + OFF_LAB);

    // zero sums + counts + hist (contiguous region)
    int zn = OFF_FN;   // 947968 floats
    WESUP_zero<<<dim3((zn + 255) / 256), dim3(256), 0, stream>>>(ws, zn);

    // streaming segment accumulation (dominant phase: 553 MB read)
    WESUP_segsum<<<dim3((CFEAT / CPB) * PCHUNKS), dim3(256), 0, stream>>>(fm, sp, sums);
    WESUP_counthist<<<dim3(HWPIX / 256), dim3(256), 0, stream>>>(sp, y, counts, hist);
    WESUP_featnorm<<<dim3(SEGS), dim3(256), 0, stream>>>(sums, counts, fn);

    // MLP via f32 WMMA (one wave per 16x16 output tile)
    // L1: feat(448x2112) @ W1(2112x1024) -> h1      28x64 tiles = 1792 waves
    WESUP_gemm_wmma<<<dim3(224), dim3(256), 0, stream>>>(sums, W1, b1, h1, 28, 64, CFEAT, 0, 0);
    // L2: h1 @ W2(1024x1024) -> h2                  28x64 tiles
    WESUP_gemm_wmma<<<dim3(224), dim3(256), 0, stream>>>(h1, W2, b2, h2, 28, 64, H1DIM, 0, 0);
    // L3: h2 @ W3(1024x32) -> h3                    28x2 tiles = 56 waves
    WESUP_gemm_wmma<<<dim3(7), dim3(256), 0, stream>>>(h2, W3, b3, h3, 28, 2, H1DIM, 0, 0);

    // pred = softmax(h3 @ Wc + bc) -> out[0..879]
    WESUP_classifier<<<dim3(2), dim3(256), 0, stream>>>(h3, Wc, bc, out);

    // labels + affinity propagation
    WESUP_labels<<<dim3(2), dim3(256), 0, stream>>>(hist, lab);
    // aff = exp(-(fn @ fn^T)) : 28x28 tiles = 784 waves, B transposed
    WESUP_gemm_wmma<<<dim3(98), dim3(256), 0, stream>>>(fn, fn, nullptr, aff, 28, 28, CFEAT, 1, 1);
    // new_lab -> out[880..1319] (as float)
    WESUP_newlab<<<dim3(SEGS), dim3(256), 0, stream>>>(aff, lab, out + 2 * SEGS);
}